// GraphSpilloverEffectEstimator_33827162423527
// MI455X (gfx1250) — compile-verified
//
#include <hip/hip_runtime.h>

#define D128 128

typedef float v2f __attribute__((ext_vector_type(2)));
typedef float v8f __attribute__((ext_vector_type(8)));

// ---------------------------------------------------------------- utilities

__global__ void zero_f32(float* __restrict__ p, int n) {
    int i = blockIdx.x * blockDim.x + threadIdx.x;
    if (i < n) p[i] = 0.0f;
}

__global__ void degree_kernel(const long long* __restrict__ ei, float* __restrict__ deg, int E) {
    int e = blockIdx.x * blockDim.x + threadIdx.x;
    if (e < E) {
        int dst = (int)ei[(long)E + e];
        atomicAdd(&deg[dst], 1.0f);
    }
}

__global__ void rsqrt_kernel(float* __restrict__ d, int n) {
    int i = blockIdx.x * blockDim.x + threadIdx.x;
    if (i < n) d[i] = rsqrtf(d[i] + 2.0f);   // improved self-loop weight 2
}

// ---------------------------------------------------------------- WMMA GEMM
// out[M x 128] = X[M x 128] @ W[128 x 128]
//              (+ X2 @ W2)  (+ g[m] * wg[n])  (+ bias[n])  (relu optional)
//
// Each wave computes a 16x64 strip = 4 independent 16x16 accumulators:
//  - one A float2 load feeds 4 WMMAs (A traffic /4)
//  - adjacent WMMAs target different accumulators -> no D->A/B RAW hazard,
//    XDL pipe stays full
// Layouts (ISA 7.12.2, f32 16x16x4):
//  A : lane l holds rows m=l&15, K-pair koff=2*(l>>4)   (2 VGPRs)
//  B : lane l holds col n=l&15, same K-pair             (2 VGPRs)
//  C : VGPR v holds row v + 8*(l>>4), col l&15          (8 VGPRs)
__global__ __launch_bounds__(256)
void gemm_wmma(const float* __restrict__ X,  const float* __restrict__ W,
               const float* __restrict__ X2, const float* __restrict__ W2,
               const float* __restrict__ g,  const float* __restrict__ wg,
               const float* __restrict__ bias, float* __restrict__ out,
               int M, int do_relu) {
    int wid  = (blockIdx.x * blockDim.x + threadIdx.x) >> 5;
    int lane = threadIdx.x & 31;
    int tilesM = M >> 4;                       // M is a multiple of 16 (50000)
    if (wid >= tilesM * 2) return;             // 2 column strips of 64
    int tm = wid >> 1, ts = wid & 1;
    int mBase = tm << 4, nBase = ts << 6;      // 0 or 64
    int r    = lane & 15;
    int koff = (lane >> 4) << 1;               // 0 or 2

    v8f acc0 = {}, acc1 = {}, acc2 = {}, acc3 = {};

    const float* xrow = X + (long)(mBase + r) * D128 + koff;
    #pragma unroll 4
    for (int k = 0; k < D128; k += 4) {
        v2f a = *(const v2f*)(xrow + k);
        const float* w0 = W + (k + koff)     * D128 + nBase + r;
        const float* w1 = W + (k + koff + 1) * D128 + nBase + r;
        v2f b0 = {w0[0],  w1[0]};
        v2f b1 = {w0[16], w1[16]};
        v2f b2 = {w0[32], w1[32]};
        v2f b3 = {w0[48], w1[48]};
        acc0 = __builtin_amdgcn_wmma_f32_16x16x4_f32(false, a, false, b0, (short)0, acc0, false, false);
        acc1 = __builtin_amdgcn_wmma_f32_16x16x4_f32(false, a, false, b1, (short)0, acc1, false, false);
        acc2 = __builtin_amdgcn_wmma_f32_16x16x4_f32(false, a, false, b2, (short)0, acc2, false, false);
        acc3 = __builtin_amdgcn_wmma_f32_16x16x4_f32(false, a, false, b3, (short)0, acc3, false, false);
    }
    if (X2) {                                  // second half of the 257-wide concat
        const float* x2row = X2 + (long)(mBase + r) * D128 + koff;
        #pragma unroll 4
        for (int k = 0; k < D128; k += 4) {
            v2f a = *(const v2f*)(x2row + k);
            const float* w0 = W2 + (k + koff)     * D128 + nBase + r;
            const float* w1 = W2 + (k + koff + 1) * D128 + nBase + r;
            v2f b0 = {w0[0],  w1[0]};
            v2f b1 = {w0[16], w1[16]};
            v2f b2 = {w0[32], w1[32]};
            v2f b3 = {w0[48], w1[48]};
            acc0 = __builtin_amdgcn_wmma_f32_16x16x4_f32(false, a, false, b0, (short)0, acc0, false, false);
            acc1 = __builtin_amdgcn_wmma_f32_16x16x4_f32(false, a, false, b1, (short)0, acc1, false, false);
            acc2 = __builtin_amdgcn_wmma_f32_16x16x4_f32(false, a, false, b2, (short)0, acc2, false, false);
            acc3 = __builtin_amdgcn_wmma_f32_16x16x4_f32(false, a, false, b3, (short)0, acc3, false, false);
        }
    }

    int mHalf = (lane >> 4) << 3;
    auto store_tile = [&](const v8f& acc, int tn) {
        int n = nBase + (tn << 4) + r;
        float wgn = wg   ? wg[n]   : 0.0f;
        float bn  = bias ? bias[n] : 0.0f;
        #pragma unroll
        for (int v = 0; v < 8; ++v) {
            int m = mBase + mHalf + v;
            float val = acc[v] + bn;
            if (g) val = fmaf(g[m], wgn, val);   // rank-1 exposure term
            if (do_relu) val = fmaxf(val, 0.0f);
            out[(long)m * D128 + n] = val;
        }
    };
    store_tile(acc0, 0);
    store_tile(acc1, 1);
    store_tile(acc2, 2);
    store_tile(acc3, 3);
}

// ---------------------------------------------------------------- edge scatter
// agg[dst] += xw[src] * dinv[src]*dinv[dst] ; one wave per edge, float4 per lane.
__global__ __launch_bounds__(256)
void scatter_edges(const long long* __restrict__ ei, const float* __restrict__ dinv,
                   const float* __restrict__ xw, float* __restrict__ agg, int E) {
    int wid  = (blockIdx.x * blockDim.x + threadIdx.x) >> 5;
    int lane = threadIdx.x & 31;
    if (wid >= E) return;
    int src = (int)ei[wid];
    int dst = (int)ei[(long)E + wid];
    float coef = dinv[src] * dinv[dst];
    const float4 x = *(const float4*)(xw + (long)src * D128 + lane * 4);
    float* o = agg + (long)dst * D128 + lane * 4;
    atomicAdd(o + 0, x.x * coef);
    atomicAdd(o + 1, x.y * coef);
    atomicAdd(o + 2, x.z * coef);
    atomicAdd(o + 3, x.w * coef);
}

// out = relu(agg + 2*dinv^2*xw + bias)
__global__ void finalize_gcn(const float* __restrict__ agg, const float* __restrict__ xw,
                             const float* __restrict__ dinv, const float* __restrict__ bias,
                             float* __restrict__ out, int total) {
    int idx = blockIdx.x * blockDim.x + threadIdx.x;
    if (idx >= total) return;
    int i = idx >> 7;
    int c = idx & 127;
    float di = dinv[i];
    float v = agg[idx] + 2.0f * di * di * xw[idx] + bias[c];
    out[idx] = fmaxf(v, 0.0f);
}

// out[i] = dot(H[i,:], w3) + b3 ; one wave per node, wave32 xor-shuffle reduction
__global__ __launch_bounds__(256)
void dot_w3(const float* __restrict__ H, const float* __restrict__ w3,
            const float* __restrict__ b3, float* __restrict__ out, int n) {
    int wid  = (blockIdx.x * blockDim.x + threadIdx.x) >> 5;
    int lane = threadIdx.x & 31;
    if (wid >= n) return;
    float4 h = *(const float4*)(H  + (long)wid * D128 + lane * 4);
    float4 w = *(const float4*)(w3 + lane * 4);
    float s = h.x * w.x + h.y * w.y + h.z * w.z + h.w * w.w;
    for (int off = 16; off > 0; off >>= 1) s += __shfl_xor(s, off, 32);
    if (lane == 0) out[wid] = s + b3[0];
}

__global__ void select_out(const int* __restrict__ tc, const float* __restrict__ t,
                           const float* __restrict__ c, float* __restrict__ out, int n) {
    int i = blockIdx.x * blockDim.x + threadIdx.x;
    if (i < n) out[i] = (tc[i] == 1) ? t[i] : c[i];
}

// ---------------------------------------------------------------- driver

extern "C" void kernel_launch(void* const* d_in, const int* in_sizes, int n_in,
                              void* d_out, int out_size, void* d_ws, size_t ws_size,
                              hipStream_t stream) {
    (void)n_in; (void)out_size; (void)ws_size;
    const float*     ind_rep = (const float*)d_in[0];
    const float*     rep     = (const float*)d_in[1];
    const int*       tc      = (const int*)d_in[2];
    const float*     g       = (const float*)d_in[3];
    const long long* ei      = (const long long*)d_in[4];
    const float* W1  = (const float*)d_in[5];   const float* b1  = (const float*)d_in[6];
    const float* W2  = (const float*)d_in[7];   const float* b2  = (const float*)d_in[8];
    const float* stW1 = (const float*)d_in[9];  const float* stb1 = (const float*)d_in[10];
    const float* stW2 = (const float*)d_in[11]; const float* stb2 = (const float*)d_in[12];
    const float* stW3 = (const float*)d_in[13]; const float* stb3 = (const float*)d_in[14];
    const float* scW1 = (const float*)d_in[15]; const float* scb1 = (const float*)d_in[16];
    const float* scW2 = (const float*)d_in[17]; const float* scb2 = (const float*)d_in[18];
    const float* scW3 = (const float*)d_in[19]; const float* scb3 = (const float*)d_in[20];

    const int N = in_sizes[2];          // 50000 (multiple of 16)
    const int E = in_sizes[4] / 2;      // 800000
    const int NDp = N * D128;

    float* out   = (float*)d_out;
    float* spill = out;                 // [N]
    float* h_rep = out + N;             // [N,128]

    float* ws   = (float*)d_ws;
    float* dinv = ws;                   // N   (deg during accumulation)
    float* T    = ws + (size_t)N;       // N
    float* C    = ws + (size_t)2 * N;   // N
    float* B0   = ws + (size_t)3 * N;   // N*128
    float* B1   = B0 + (size_t)NDp;     // N*128
    float* B2   = B1 + (size_t)NDp;     // N*128

    const int TB = 256;
    int blkN    = (N + TB - 1) / TB;
    int blkE    = (E + TB - 1) / TB;
    int blkND   = (NDp + TB - 1) / TB;
    int blkGemm = ((N / 16) * 2 * 32 + TB - 1) / TB;   // one wave per 16x64 strip
    int blkEw   = (E * 32 + TB - 1) / TB;              // one wave per edge
    int blkNw   = (N * 32 + TB - 1) / TB;              // one wave per node

    // degrees -> dinv
    zero_f32<<<blkN, TB, 0, stream>>>(dinv, N);
    degree_kernel<<<blkE, TB, 0, stream>>>(ei, dinv, E);
    rsqrt_kernel<<<blkN, TB, 0, stream>>>(dinv, N);

    // GCN layer 1: xw -> scatter -> relu
    gemm_wmma<<<blkGemm, TB, 0, stream>>>(rep, W1, nullptr, nullptr, nullptr, nullptr,
                                          nullptr, B0, N, 0);
    zero_f32<<<blkND, TB, 0, stream>>>(B1, NDp);
    scatter_edges<<<blkEw, TB, 0, stream>>>(ei, dinv, B0, B1, E);
    finalize_gcn<<<blkND, TB, 0, stream>>>(B1, B0, dinv, b1, B2, NDp);

    // GCN layer 2 -> h_rep (second output)
    gemm_wmma<<<blkGemm, TB, 0, stream>>>(B2, W2, nullptr, nullptr, nullptr, nullptr,
                                          nullptr, B0, N, 0);
    zero_f32<<<blkND, TB, 0, stream>>>(B1, NDp);
    scatter_edges<<<blkEw, TB, 0, stream>>>(ei, dinv, B0, B1, E);
    finalize_gcn<<<blkND, TB, 0, stream>>>(B1, B0, dinv, b2, h_rep, NDp);

    // spill_t MLP on [ind_rep | h_rep | g]
    gemm_wmma<<<blkGemm, TB, 0, stream>>>(ind_rep, stW1, h_rep, stW1 + 128 * D128,
                                          g, stW1 + 256 * D128, stb1, B0, N, 1);
    gemm_wmma<<<blkGemm, TB, 0, stream>>>(B0, stW2, nullptr, nullptr, nullptr, nullptr,
                                          stb2, B1, N, 1);
    dot_w3<<<blkNw, TB, 0, stream>>>(B1, stW3, stb3, T, N);

    // spill_c MLP
    gemm_wmma<<<blkGemm, TB, 0, stream>>>(ind_rep, scW1, h_rep, scW1 + 128 * D128,
                                          g, scW1 + 256 * D128, scb1, B0, N, 1);
    gemm_wmma<<<blkGemm, TB, 0, stream>>>(B0, scW2, nullptr, nullptr, nullptr, nullptr,
                                          scb2, B1, N, 1);
    dot_w3<<<blkNw, TB, 0, stream>>>(B1, scW3, scb3, C, N);

    // treatment/control select
    select_out<<<blkN, TB, 0, stream>>>(tc, T, C, spill, N);
}